// SimpleGraphConv_21320217658122
// MI455X (gfx1250) — compile-verified
//
#include <hip/hip_runtime.h>

typedef __attribute__((ext_vector_type(2))) float v2f;
typedef __attribute__((ext_vector_type(8))) float v8f;
typedef __attribute__((ext_vector_type(4))) int   v4i;
typedef __attribute__((address_space(1))) v4i     gas_v4i;  // global
typedef __attribute__((address_space(3))) v4i     las_v4i;  // LDS

#define N_NODES 50000
#define N_EDGES 800000
#define D 128
#define ROWS_PB 64
#define LDS_STRIDE 132  // pad 128 -> 132 floats: A-frag LDS reads hit distinct banks

#if __has_builtin(__builtin_amdgcn_global_load_async_to_lds_b128)
#define HAVE_ASYNC_LDS 1
#else
#define HAVE_ASYNC_LDS 0
#endif

// ---------------- zero the SpMM accumulator ----------------
__global__ void k_zero(float* __restrict__ acc, int n4) {
  int i = blockIdx.x * blockDim.x + threadIdx.x;
  if (i < n4) ((float4*)acc)[i] = make_float4(0.f, 0.f, 0.f, 0.f);
}

// ---------------- dense GEMM: HW = H @ W via V_WMMA_F32_16X16X4_F32 ----------------
__global__ __launch_bounds__(256)
void k_gemm(const float* __restrict__ H, const float* __restrict__ W,
            float* __restrict__ HW) {
  __shared__ float Hs[ROWS_PB * LDS_STRIDE];

  const int tid  = threadIdx.x;
  const int lane = tid & 31;
  const int wid  = tid >> 5;             // 0..7: column tile of W
  const int rowbase = blockIdx.x * ROWS_PB;

  // Stage 64x128 strip of H into LDS.
#if HAVE_ASYNC_LDS
  // gfx1250 async path: GLOBAL_LOAD_ASYNC_TO_LDS_B128, no VGPR round trip.
  for (int i = tid; i < ROWS_PB * (D / 4); i += 256) {
    int r  = i / (D / 4);
    int c4 = (i % (D / 4)) * 4;
    int row = rowbase + r;
    row = row < N_NODES ? row : N_NODES - 1;   // clamped tail read
    __builtin_amdgcn_global_load_async_to_lds_b128(
        (gas_v4i*)(H + (size_t)row * D + c4),
        (las_v4i*)&Hs[r * LDS_STRIDE + c4],
        /*offset=*/0, /*cpol=*/0);
  }
#if __has_builtin(__builtin_amdgcn_s_wait_asynccnt)
  __builtin_amdgcn_s_wait_asynccnt(0);
#else
  asm volatile("s_wait_asynccnt 0" ::: "memory");
#endif
  __syncthreads();
#else
  for (int i = tid; i < ROWS_PB * (D / 4); i += 256) {
    int r  = i / (D / 4);
    int c4 = (i % (D / 4)) * 4;
    int row = rowbase + r;
    row = row < N_NODES ? row : N_NODES - 1;
    const float4 v = *(const float4*)(H + (size_t)row * D + c4);
    float* dst = &Hs[r * LDS_STRIDE + c4];
    dst[0] = v.x; dst[1] = v.y; dst[2] = v.z; dst[3] = v.w;
  }
  __syncthreads();
#endif

  // Preload all 32 B fragments for this wave's 16-col slice of W.
  // B (4x16, f32, 2 VGPRs): K = vgpr + 2*(lane>>4), N = lane&15.
  const int nb    = wid * 16 + (lane & 15);
  const int khalf = (lane >> 4) * 2;
  v2f b[32];
#pragma unroll
  for (int s = 0; s < 32; ++s) {
    int k = 4 * s + khalf;
    b[s].x = W[(size_t)k * D + nb];
    b[s].y = W[(size_t)(k + 1) * D + nb];
  }

  v8f acc[4];
  // A (16x4, f32, 2 VGPRs): M = lane&15, K = 2*(lane>>4) + j.
#pragma unroll
  for (int rt = 0; rt < 4; ++rt) {
    v8f c = {};
    const float* arow = &Hs[(rt * 16 + (lane & 15)) * LDS_STRIDE + khalf];
#pragma unroll
    for (int s = 0; s < 32; ++s) {
      v2f a;
      a.x = arow[4 * s];
      a.y = arow[4 * s + 1];
      c = __builtin_amdgcn_wmma_f32_16x16x4_f32(
          /*neg_a=*/false, a, /*neg_b=*/false, b[s],
          /*c_mod=*/(short)0, c, /*reuse_a=*/false, /*reuse_b=*/false);
    }
    acc[rt] = c;
  }

  // D (16x16, f32, 8 VGPRs): M = vgpr + 8*(lane>>4), N = lane&15.
  const int n = wid * 16 + (lane & 15);
  if (rowbase + ROWS_PB <= N_NODES) {
    // block-uniform fast path: no per-lane predication
#pragma unroll
    for (int rt = 0; rt < 4; ++rt) {
#pragma unroll
      for (int i = 0; i < 8; ++i) {
        int m = rowbase + rt * 16 + i + 8 * (lane >> 4);
        HW[(size_t)m * D + n] = acc[rt][i];
      }
    }
  } else {
#pragma unroll
    for (int rt = 0; rt < 4; ++rt) {
#pragma unroll
      for (int i = 0; i < 8; ++i) {
        int m = rowbase + rt * 16 + i + 8 * (lane >> 4);
        if (m < N_NODES) HW[(size_t)m * D + n] = acc[rt][i];
      }
    }
  }
}

// ---------------- COO SpMM scatter: one wave32 per edge ----------------
__global__ __launch_bounds__(256)
void k_spmm(const float* __restrict__ HW, const float* __restrict__ vals,
            const int* __restrict__ rows, const int* __restrict__ cols,
            float* __restrict__ acc) {
  int wave = (blockIdx.x * blockDim.x + threadIdx.x) >> 5;
  int lane = threadIdx.x & 31;
  if (wave >= N_EDGES) return;
  int   r = rows[wave];
  int   c = cols[wave];
  float v = vals[wave];
  // 128-float row = 32 lanes x float4: fully coalesced, L2-resident gather
  float4 h = *(const float4*)(HW + (size_t)c * D + lane * 4);
  float* dst = acc + (size_t)r * D + lane * 4;
  __hip_atomic_fetch_add(dst + 0, h.x * v, __ATOMIC_RELAXED, __HIP_MEMORY_SCOPE_AGENT);
  __hip_atomic_fetch_add(dst + 1, h.y * v, __ATOMIC_RELAXED, __HIP_MEMORY_SCOPE_AGENT);
  __hip_atomic_fetch_add(dst + 2, h.z * v, __ATOMIC_RELAXED, __HIP_MEMORY_SCOPE_AGENT);
  __hip_atomic_fetch_add(dst + 3, h.w * v, __ATOMIC_RELAXED, __HIP_MEMORY_SCOPE_AGENT);
}

// ---------------- bias + ReLU into d_out ----------------
__global__ void k_bias_relu(const float* __restrict__ acc, const float* __restrict__ bias,
                            float* __restrict__ out, int n4) {
  int i = blockIdx.x * blockDim.x + threadIdx.x;
  if (i >= n4) return;
  float4 a = ((const float4*)acc)[i];
  int cb = (i * 4) & (D - 1);
  float4 bv = *(const float4*)(bias + cb);
  float4 o;
  o.x = fmaxf(a.x + bv.x, 0.f);
  o.y = fmaxf(a.y + bv.y, 0.f);
  o.z = fmaxf(a.z + bv.z, 0.f);
  o.w = fmaxf(a.w + bv.w, 0.f);
  ((float4*)out)[i] = o;
}

extern "C" void kernel_launch(void* const* d_in, const int* in_sizes, int n_in,
                              void* d_out, int out_size, void* d_ws, size_t ws_size,
                              hipStream_t stream) {
  const float* H     = (const float*)d_in[0];
  const float* W     = (const float*)d_in[1];
  const float* bias  = (const float*)d_in[2];
  const float* evals = (const float*)d_in[3];
  const int*   erows = (const int*)d_in[4];
  const int*   ecols = (const int*)d_in[5];
  float* out = (float*)d_out;

  float* HW  = (float*)d_ws;                     // 50000*128 f32 = 25.6 MB
  float* ACC = HW + (size_t)N_NODES * D;         // 50000*128 f32 = 25.6 MB

  const int n4 = N_NODES * D / 4;
  k_zero<<<(n4 + 255) / 256, 256, 0, stream>>>(ACC, n4);
  k_gemm<<<(N_NODES + ROWS_PB - 1) / ROWS_PB, 256, 0, stream>>>(H, W, HW);
  k_spmm<<<(N_EDGES * 32) / 256, 256, 0, stream>>>(HW, evals, erows, ecols, ACC);
  k_bias_relu<<<(n4 + 255) / 256, 256, 0, stream>>>(ACC, bias, out, n4);
}